// MedianTripletHead_30185030156997
// MI455X (gfx1250) — compile-verified
//
#include <hip/hip_runtime.h>
#include <hip/hip_bf16.h>

typedef __attribute__((ext_vector_type(2))) float v2f;
typedef __attribute__((ext_vector_type(8))) float v8f;

static constexpr int kN = 4096;      // rows (and output dim)
static constexpr int kC = 2048;      // feature dim
static constexpr float kGamma  = 2.0f;
static constexpr float kMargin = 2.0f;
static constexpr float kEps    = 1e-12f;

// ---------------------------------------------------------------------------
// Kernel 1: per-row inverse L2 norms for both input and target.
// blockIdx.x in [0, 2N): first N rows -> invA (input), rest -> invB (target).
// ---------------------------------------------------------------------------
__global__ void __launch_bounds__(256) rownorm_kernel(const float* __restrict__ in,
                                                      const float* __restrict__ tg,
                                                      float* __restrict__ invA,
                                                      float* __restrict__ invB) {
    int b = blockIdx.x;
    const float* src;
    float* dst;
    int row;
    if (b < kN) { src = in; dst = invA; row = b; }
    else        { src = tg; dst = invB; row = b - kN; }

    const float* p = src + (size_t)row * kC;
    float s = 0.0f;
    for (int j = threadIdx.x; j < kC; j += 256) {
        float v = p[j];
        s += v * v;
    }
    __shared__ float red[256];
    red[threadIdx.x] = s;
    __syncthreads();
    for (int off = 128; off > 0; off >>= 1) {
        if (threadIdx.x < off) red[threadIdx.x] += red[threadIdx.x + off];
        __syncthreads();
    }
    if (threadIdx.x == 0) {
        float nrm = sqrtf(red[0]);
        dst[row] = 1.0f / fmaxf(nrm, kEps);
    }
}

// ---------------------------------------------------------------------------
// Kernel 2: dist[i,j] = -(A_i . T_j) * invA[i] * invB[j] via V_WMMA_F32_16X16X4_F32.
// One wave32 computes a 16-row x 64-col strip (4 accumulator tiles, A reused).
//
// Fragment layout (ISA 7.12.2, fp32):
//   A (16x4,  MxK): lane 0-15 -> M=lane,    VGPRs hold K={0,1}; lanes 16-31 K={2,3}
//   B (4x16,  KxN): lane 0-15 -> N=lane,    VGPRs hold K={0,1}; lanes 16-31 K={2,3}
//   C/D (16x16):    VGPR r: lanes 0-15 -> M=r, N=lane; lanes 16-31 -> M=8+r, N=lane-16
// Both A and B fragments are contiguous float2 loads from row-major memory
// (B's K-major-per-lane layout matches target rows acting as B columns).
// ---------------------------------------------------------------------------
__global__ void __launch_bounds__(32) gemm_wmma_kernel(const float* __restrict__ A,
                                                       const float* __restrict__ T,
                                                       const float* __restrict__ invA,
                                                       const float* __restrict__ invB,
                                                       float* __restrict__ dist) {
    const int lane = threadIdx.x;        // 0..31
    const int lid  = lane & 15;
    const int half = lane >> 4;          // 0 or 1
    const int m0   = blockIdx.y * 16;    // output row tile
    const int n0   = blockIdx.x * 64;    // output col strip

    const float* aRow = A + (size_t)(m0 + lid) * kC + 2 * half;
    const float* bR0  = T + (size_t)(n0 +  0 + lid) * kC + 2 * half;
    const float* bR1  = T + (size_t)(n0 + 16 + lid) * kC + 2 * half;
    const float* bR2  = T + (size_t)(n0 + 32 + lid) * kC + 2 * half;
    const float* bR3  = T + (size_t)(n0 + 48 + lid) * kC + 2 * half;

    v8f acc0 = {}, acc1 = {}, acc2 = {}, acc3 = {};

    for (int k0 = 0; k0 < kC; k0 += 4) {
        v2f a  = *(const v2f*)(aRow + k0);
        v2f x0 = *(const v2f*)(bR0 + k0);
        v2f x1 = *(const v2f*)(bR1 + k0);
        v2f x2 = *(const v2f*)(bR2 + k0);
        v2f x3 = *(const v2f*)(bR3 + k0);
        // 8 args: (neg_a, A, neg_b, B, c_mod, C, reuse_a, reuse_b)
        acc0 = __builtin_amdgcn_wmma_f32_16x16x4_f32(false, a, false, x0, (short)0, acc0, false, false);
        acc1 = __builtin_amdgcn_wmma_f32_16x16x4_f32(false, a, false, x1, (short)0, acc1, false, false);
        acc2 = __builtin_amdgcn_wmma_f32_16x16x4_f32(false, a, false, x2, (short)0, acc2, false, false);
        acc3 = __builtin_amdgcn_wmma_f32_16x16x4_f32(false, a, false, x3, (short)0, acc3, false, false);
    }

    const float ib0 = invB[n0 +  0 + lid];
    const float ib1 = invB[n0 + 16 + lid];
    const float ib2 = invB[n0 + 32 + lid];
    const float ib3 = invB[n0 + 48 + lid];

#pragma unroll
    for (int r = 0; r < 8; ++r) {
        const int row = m0 + r + 8 * half;
        const float ia = invA[row];
        float* drow = dist + (size_t)row * kN;
        drow[n0 +  0 + lid] = -(acc0[r] * ia * ib0);
        drow[n0 + 16 + lid] = -(acc1[r] * ia * ib1);
        drow[n0 + 32 + lid] = -(acc2[r] * ia * ib2);
        drow[n0 + 48 + lid] = -(acc3[r] * ia * ib3);
    }
}

// ---------------------------------------------------------------------------
// Kernel 3: per-row exact radix-select median of the off-diagonal elements,
// plus diagonal extraction. One 256-thread block per row; row cached in LDS
// as order-preserving uint keys; diagonal poisoned to 0xFFFFFFFF (max key).
// k = (N-2)/2 = 2047 == torch lower-median index of the 4095 off-diag elems.
// ---------------------------------------------------------------------------
__global__ void __launch_bounds__(256) median_kernel(const float* __restrict__ dist,
                                                     float* __restrict__ an,
                                                     float* __restrict__ ap) {
    const int row = blockIdx.x;
    const int tid = threadIdx.x;

    __shared__ unsigned keys[kN];
    __shared__ unsigned s_cnt;
    __shared__ float s_ap;

    const float* drow = dist + (size_t)row * kN;
    for (int j = tid; j < kN; j += 256) {
        float f = drow[j];
        unsigned key;
        if (j == row) {
            s_ap = f;
            key = 0xFFFFFFFFu;   // push diagonal to +inf end
        } else {
            unsigned bits = __float_as_uint(f);
            key = (bits & 0x80000000u) ? ~bits : (bits | 0x80000000u);
        }
        keys[j] = key;
    }
    __syncthreads();

    unsigned prefix = 0u;
    unsigned want = (unsigned)((kN - 2) / 2);   // 2047

    for (int bit = 31; bit >= 0; --bit) {
        if (tid == 0) s_cnt = 0u;
        __syncthreads();

        const unsigned himask = (bit == 31) ? 0u : (0xFFFFFFFFu << (bit + 1));
        unsigned local = 0u;
        for (int j = tid; j < kN; j += 256) {
            unsigned key = keys[j];
            if ((key & himask) == prefix && ((key >> bit) & 1u) == 0u) local++;
        }
        atomicAdd(&s_cnt, local);
        __syncthreads();

        const unsigned c = s_cnt;
        if (want >= c) {
            prefix |= (1u << bit);
            want -= c;
        }
        __syncthreads();   // protect s_cnt reset on next iteration
    }

    if (tid == 0) {
        unsigned bits = (prefix & 0x80000000u) ? (prefix & 0x7FFFFFFFu) : ~prefix;
        an[row] = __uint_as_float(bits);
        ap[row] = s_ap;
    }
}

// ---------------------------------------------------------------------------
// Kernel 4: loss = mean(relu(gamma*ap - an + margin))
// ---------------------------------------------------------------------------
__global__ void __launch_bounds__(256) loss_kernel(const float* __restrict__ ap,
                                                   const float* __restrict__ an,
                                                   float* __restrict__ out) {
    __shared__ float red[256];
    float s = 0.0f;
    for (int i = threadIdx.x; i < kN; i += 256) {
        float v = kGamma * ap[i] - an[i] + kMargin;
        s += fmaxf(v, 0.0f);
    }
    red[threadIdx.x] = s;
    __syncthreads();
    for (int off = 128; off > 0; off >>= 1) {
        if (threadIdx.x < off) red[threadIdx.x] += red[threadIdx.x + off];
        __syncthreads();
    }
    if (threadIdx.x == 0) out[0] = red[0] / (float)kN;
}

// ---------------------------------------------------------------------------
extern "C" void kernel_launch(void* const* d_in, const int* in_sizes, int n_in,
                              void* d_out, int out_size, void* d_ws, size_t ws_size,
                              hipStream_t stream) {
    const float* input  = (const float*)d_in[0];   // [4096, 2048] fp32
    const float* target = (const float*)d_in[1];   // [4096, 2048] fp32

    float* ws   = (float*)d_ws;
    float* invA = ws;                 // [N]
    float* invB = ws + kN;            // [N]
    float* ap   = ws + 2 * kN;        // [N]
    float* an   = ws + 3 * kN;        // [N]
    float* dist = ws + 4 * kN;        // [N, N]  (64 MB)

    rownorm_kernel<<<2 * kN, 256, 0, stream>>>(input, target, invA, invB);

    dim3 ggrid(kN / 64, kN / 16);     // (64, 256)
    gemm_wmma_kernel<<<ggrid, 32, 0, stream>>>(input, target, invA, invB, dist);

    median_kernel<<<kN, 256, 0, stream>>>(dist, an, ap);

    loss_kernel<<<1, 256, 0, stream>>>(ap, an, (float*)d_out);
}